// AttnTopKPool_66082366816340
// MI455X (gfx1250) — compile-verified
//
#include <hip/hip_runtime.h>

typedef __attribute__((ext_vector_type(2))) float v2f;
typedef __attribute__((ext_vector_type(4))) float v4f;
typedef __attribute__((ext_vector_type(8))) float v8f;

#define BATCH 32
#define FDIM  1024
#define SDIM  2048
#define KTOP  16
#define RCHUNKS 16
#define ROWS_PER_CHUNK (SDIM / RCHUNKS)   // 128

// Four-tile WMMA step: q0/q1 hold one row each (4 cols per lane); component t
// feeds tile t. A = ones(16x4) -> D[m,n] = sum_k B[k,n], K-mapping irrelevant.
__device__ __forceinline__ void wmma_step(const v2f& a, v4f q0, v4f q1,
                                          v8f& c0, v8f& c1, v8f& c2, v8f& c3) {
  v2f b0, b1, b2, b3;
  b0[0] = q0[0]; b0[1] = q1[0];
  b1[0] = q0[1]; b1[1] = q1[1];
  b2[0] = q0[2]; b2[1] = q1[2];
  b3[0] = q0[3]; b3[1] = q1[3];
  c0 = __builtin_amdgcn_wmma_f32_16x16x4_f32(false, a, false, b0,
                                             (short)0, c0, false, false);
  c1 = __builtin_amdgcn_wmma_f32_16x16x4_f32(false, a, false, b1,
                                             (short)0, c1, false, false);
  c2 = __builtin_amdgcn_wmma_f32_16x16x4_f32(false, a, false, b2,
                                             (short)0, c2, false, false);
  c3 = __builtin_amdgcn_wmma_f32_16x16x4_f32(false, a, false, b3,
                                             (short)0, c3, false, false);
}

// ---------------------------------------------------------------------------
// Kernel 1: column partial sums of w via V_WMMA_F32_16X16X4_F32.
// Each wave owns 64 columns = 4 interleaved 16-wide tiles (tile t = columns
// col0 + 4n + t). Per-lane b128 loads give 256 B contiguous per half-wave (two rows per wave-load);
// lanes 0-15 supply rows r+0/r+1, lanes 16-31 rows r+2/r+3. Software-
// pipelined double buffer keeps next iteration's loads in flight during the
// current shuffle+WMMA. Result: c_t[0] lane n -> column col0+4n+t, so the
// store is a contiguous b128 per lane.
// Grid: BATCH * 4(colGroups of 512 cols) * RCHUNKS blocks, 256 thr = 8 waves.
// ---------------------------------------------------------------------------
__global__ __launch_bounds__(256) void colsum_wmma(
    const float* __restrict__ w, float* __restrict__ partial) {
  const int lane = threadIdx.x & 31;
  const int wv   = threadIdx.x >> 5;            // wave in block: 0..7
  const int blk  = blockIdx.x;
  const int b      = blk >> 6;                  // / (4*16)
  const int rem    = blk & 63;
  const int cgrp   = rem >> 4;                  // 0..3
  const int rchunk = rem & 15;                  // 0..15

  const int col0  = (cgrp * 8 + wv) * 64;       // 64-column tile base
  const int n     = lane & 15;                  // column-slot within tiles
  const int krow0 = (lane >> 4) * 2;            // 0 (lanes 0-15) or 2 (16-31)
  const int rbase = rchunk * ROWS_PER_CHUNK;

  const float* base = w + (size_t)b * SDIM * SDIM + (size_t)rbase * SDIM
                        + col0 + 4 * n + (size_t)krow0 * SDIM;

  v2f a; a[0] = 1.0f; a[1] = 1.0f;              // ones A-matrix (16x4)
  v8f c0 = {}, c1 = {}, c2 = {}, c3 = {};       // 4 independent accumulators

  // Prologue: first row pair in flight.
  v4f q0 = *(const v4f*)(base);
  v4f q1 = *(const v4f*)(base + (size_t)SDIM);

#pragma unroll 2
  for (int r = 0; r < ROWS_PER_CHUNK - 4; r += 4) {
    const float* pn = base + (size_t)(r + 4) * SDIM;
    v4f n0 = *(const v4f*)(pn);                 // next iteration's rows
    v4f n1 = *(const v4f*)(pn + (size_t)SDIM);
    if (r + 20 < ROWS_PER_CHUNK)
      __builtin_prefetch(pn + (size_t)12 * SDIM, 0, 1);  // global_prefetch_b8

    wmma_step(a, q0, q1, c0, c1, c2, c3);       // consume current buffer
    q0 = n0; q1 = n1;                           // rotate
  }
  wmma_step(a, q0, q1, c0, c1, c2, c3);         // epilogue

  // Contiguous b128 store: lane n -> columns col0+4n .. col0+4n+3.
  if (lane < 16) {
    v4f s;
    s[0] = c0[0]; s[1] = c1[0]; s[2] = c2[0]; s[3] = c3[0];
    *(v4f*)(partial + ((size_t)rchunk * BATCH + b) * SDIM + col0 + 4 * n) = s;
  }
}

// ---------------------------------------------------------------------------
// Kernel 2: per-batch top-16 (descending value, ties -> smaller index, like
// lax.top_k). One block per batch; folds RCHUNKS partials in fixed order
// (bit-deterministic across replays).
// ---------------------------------------------------------------------------
__global__ __launch_bounds__(256) void topk_kernel(
    const float* __restrict__ partial, int* __restrict__ idx) {
  __shared__ float vals[SDIM];
  __shared__ float rv[256];
  __shared__ int   ri[256];
  const int b = blockIdx.x;
  const int t = threadIdx.x;

  for (int j = t; j < SDIM; j += 256) {
    float s = 0.0f;
    for (int rc = 0; rc < RCHUNKS; ++rc)
      s += partial[((size_t)rc * BATCH + b) * SDIM + j];
    vals[j] = s;                                // sum ~ mean (positive scale)
  }
  __syncthreads();

  for (int k = 0; k < KTOP; ++k) {
    float best = -__builtin_huge_valf();
    int   bi   = 0x7fffffff;
    for (int j = t; j < SDIM; j += 256) {
      float v = vals[j];
      if (v > best || (v == best && j < bi)) { best = v; bi = j; }
    }
    rv[t] = best; ri[t] = bi;
    __syncthreads();
    for (int off = 128; off > 0; off >>= 1) {
      if (t < off) {
        float v2 = rv[t + off]; int i2 = ri[t + off];
        if (v2 > rv[t] || (v2 == rv[t] && i2 < ri[t])) { rv[t] = v2; ri[t] = i2; }
      }
      __syncthreads();
    }
    int sel = ri[0];
    if (t == 0) {
      idx[b * KTOP + k] = sel;
      vals[sel] = -__builtin_huge_valf();
    }
    __syncthreads();
  }
}

// ---------------------------------------------------------------------------
// Kernel 3: out[b,f,k] = x[b,f,idx[b,k]]; coalesced writes, gather reads.
// ---------------------------------------------------------------------------
__global__ __launch_bounds__(256) void gather_kernel(
    const float* __restrict__ x, const int* __restrict__ idx,
    float* __restrict__ out) {
  const int o = blockIdx.x * 256 + threadIdx.x;
  const int k = o & (KTOP - 1);
  const int f = (o >> 4) & (FDIM - 1);
  const int b = o >> 14;                        // FDIM*KTOP = 16384
  const int col = idx[b * KTOP + k];
  out[o] = x[((size_t)b * FDIM + f) * SDIM + col];
}

// ---------------------------------------------------------------------------
extern "C" void kernel_launch(void* const* d_in, const int* in_sizes, int n_in,
                              void* d_out, int out_size, void* d_ws,
                              size_t ws_size, hipStream_t stream) {
  const float* x = (const float*)d_in[0];   // [B, F, S] f32
  const float* w = (const float*)d_in[1];   // [B, S, S] f32
  float* out = (float*)d_out;               // [B, F, K] f32

  float* partial = (float*)d_ws;            // RCHUNKS*BATCH*SDIM f32 = 4 MiB
  int*   idx = (int*)((char*)d_ws +
                      (size_t)RCHUNKS * BATCH * SDIM * sizeof(float));

  colsum_wmma<<<dim3(BATCH * 4 * RCHUNKS), dim3(256), 0, stream>>>(w, partial);
  topk_kernel<<<dim3(BATCH), dim3(256), 0, stream>>>(partial, idx);

  const int total = BATCH * FDIM * KTOP;    // 524288
  gather_kernel<<<dim3(total / 256), dim3(256), 0, stream>>>(x, idx, out);
}